// MPNN_28114855919905
// MI455X (gfx1250) — compile-verified
//
#include <hip/hip_runtime.h>

// MPNN on MI455X (gfx1250, wave32, WMMA).
// All GEMMs via V_WMMA_F32_16X16X4_F32 (exact fp32 vs reference).
// Edge encoder + edge MLP + scatter-add fused into one kernel to keep HBM
// traffic near the 250MB floor (ef write dominates); nf gather table (12.8MB)
// stays resident in the 192MB L2.

#define N_NODES 50000
#define N_EDGES 800000
#define DMODEL  64

typedef __attribute__((ext_vector_type(2))) float v2f;
typedef __attribute__((ext_vector_type(8))) float v8f;

__device__ __forceinline__ v8f wmma_f32(v2f a, v2f b, v8f c) {
  // 8 args: (neg_a, A, neg_b, B, c_mod, C, reuse_a, reuse_b)
  return __builtin_amdgcn_wmma_f32_16x16x4_f32(false, a, false, b, (short)0, c,
                                               false, false);
}

__global__ void zero_kernel(float* __restrict__ p, int n) {
  int i = blockIdx.x * blockDim.x + threadIdx.x;
  if (i < n) p[i] = 0.0f;
}

// nf_enc[N,64] = x[N,16] @ W[16,64] + b  — one wave per 16-row tile, 4 col tiles.
__global__ void __launch_bounds__(256)
node_enc_kernel(const float* __restrict__ x, const float* __restrict__ W,
                const float* __restrict__ bias, float* __restrict__ nf) {
  int wave = (blockIdx.x * blockDim.x + threadIdx.x) >> 5;
  if (wave >= N_NODES / 16) return;           // wave-uniform: EXEC stays full
  int lane = threadIdx.x & 31;
  int h = lane >> 4, m = lane & 15;
  int row0 = wave * 16;
  const float* xrow = x + (row0 + m) * 16;
  v8f acc[4] = {};
  for (int k = 0; k < 16; k += 4) {
    int kk = k + 2 * h;
    v2f a; a.x = xrow[kk]; a.y = xrow[kk + 1];
#pragma unroll
    for (int c = 0; c < 4; ++c) {
      v2f b;
      b.x = W[kk * DMODEL + c * 16 + m];
      b.y = W[(kk + 1) * DMODEL + c * 16 + m];
      acc[c] = wmma_f32(a, b, acc[c]);
    }
  }
#pragma unroll
  for (int c = 0; c < 4; ++c) {
    int col = c * 16 + m;
    float bv = bias[col];
#pragma unroll
    for (int r = 0; r < 8; ++r)
      nf[(row0 + r + 8 * h) * DMODEL + col] = acc[c][r] + bv;
  }
}

// Fused: ef_enc = ea@Wee+bee (WMMA -> LDS);
//        ef_new = ef_enc + relu([nf[src]|nf[dst]|ef_enc] @ Wem + bem);
//        write ef_new; atomicAdd into agg[dst].
__global__ void __launch_bounds__(256)
edge_fused_kernel(const float* __restrict__ ea, const float* __restrict__ Wee,
                  const float* __restrict__ bee, const float* __restrict__ nf,
                  const float* __restrict__ Wem, const float* __restrict__ bem,
                  const int* __restrict__ srcI, const int* __restrict__ dstI,
                  float* __restrict__ ef_out, float* __restrict__ agg) {
  __shared__ float lds[8][16 * DMODEL];       // 4KB per wave (ef_enc tile)
  int warp = threadIdx.x >> 5;
  int lane = threadIdx.x & 31;
  int h = lane >> 4, m = lane & 15;
  int tile = blockIdx.x * 8 + warp;           // grid sized exactly: 50000 tiles
  int row0 = tile * 16;
  float* sl = &lds[warp][0];

  // ---- stage 1: edge encoder tile (K=8) into LDS ----
  {
    v8f acc[4] = {};
    const float* arow = ea + (row0 + m) * 8;
#pragma unroll
    for (int k = 0; k < 8; k += 4) {
      int kk = k + 2 * h;
      v2f a; a.x = arow[kk]; a.y = arow[kk + 1];
#pragma unroll
      for (int c = 0; c < 4; ++c) {
        v2f b;
        b.x = Wee[kk * DMODEL + c * 16 + m];
        b.y = Wee[(kk + 1) * DMODEL + c * 16 + m];
        acc[c] = wmma_f32(a, b, acc[c]);
      }
    }
#pragma unroll
    for (int c = 0; c < 4; ++c) {
      int col = c * 16 + m;
      float bv = bee[col];
#pragma unroll
      for (int r = 0; r < 8; ++r)
        sl[(r + 8 * h) * DMODEL + col] = acc[c][r] + bv;
    }
  }
  __builtin_amdgcn_wave_barrier();  // DS ops are in-order per wave; fence sched

  // ---- stage 2: edge MLP, K = 192 = [nf[src] | nf[dst] | ef_enc] ----
  int e = row0 + m;
  const float* srow = nf + srcI[e] * DMODEL;  // L2-resident gather
  const float* drow = nf + dstI[e] * DMODEL;

  v8f acc[4] = {};
  for (int k = 0; k < 64; k += 4) {           // K block 0..63: nf[src]
    int kk = k + 2 * h;
    v2f a; a.x = srow[kk]; a.y = srow[kk + 1];
#pragma unroll
    for (int c = 0; c < 4; ++c) {
      v2f b;
      b.x = Wem[kk * DMODEL + c * 16 + m];
      b.y = Wem[(kk + 1) * DMODEL + c * 16 + m];
      acc[c] = wmma_f32(a, b, acc[c]);
    }
  }
  for (int k = 0; k < 64; k += 4) {           // K block 64..127: nf[dst]
    int kk = k + 2 * h;
    v2f a; a.x = drow[kk]; a.y = drow[kk + 1];
#pragma unroll
    for (int c = 0; c < 4; ++c) {
      int wr = 64 + kk;
      v2f b;
      b.x = Wem[wr * DMODEL + c * 16 + m];
      b.y = Wem[(wr + 1) * DMODEL + c * 16 + m];
      acc[c] = wmma_f32(a, b, acc[c]);
    }
  }
  for (int k = 0; k < 64; k += 4) {           // K block 128..191: ef_enc (LDS)
    int kk = k + 2 * h;
    v2f a; a.x = sl[m * DMODEL + kk]; a.y = sl[m * DMODEL + kk + 1];
#pragma unroll
    for (int c = 0; c < 4; ++c) {
      int wr = 128 + kk;
      v2f b;
      b.x = Wem[wr * DMODEL + c * 16 + m];
      b.y = Wem[(wr + 1) * DMODEL + c * 16 + m];
      acc[c] = wmma_f32(a, b, acc[c]);
    }
  }

  // ---- epilogue: residual + relu, store ef, scatter-add into agg ----
  int dsts[8];
#pragma unroll
  for (int r = 0; r < 8; ++r) dsts[r] = dstI[row0 + r + 8 * h];
#pragma unroll
  for (int c = 0; c < 4; ++c) {
    int col = c * 16 + m;
    float bv = bem[col];
#pragma unroll
    for (int r = 0; r < 8; ++r) {
      int row = r + 8 * h;
      float v = acc[c][r] + bv;
      v = v > 0.0f ? v : 0.0f;
      v += sl[row * DMODEL + col];            // residual (ef_enc)
      ef_out[(row0 + row) * DMODEL + col] = v;
      unsafeAtomicAdd(&agg[dsts[r] * DMODEL + col], v);  // global_atomic_add_f32
    }
  }
}

// nf_out = nf + relu([nf | agg] @ W[128,64] + b)
__global__ void __launch_bounds__(256)
node_mlp_kernel(const float* __restrict__ nf, const float* __restrict__ agg,
                const float* __restrict__ W, const float* __restrict__ bias,
                float* __restrict__ out) {
  int wave = (blockIdx.x * blockDim.x + threadIdx.x) >> 5;
  if (wave >= N_NODES / 16) return;
  int lane = threadIdx.x & 31;
  int h = lane >> 4, m = lane & 15;
  int row0 = wave * 16;
  const float* arow = nf + (row0 + m) * DMODEL;
  const float* grow = agg + (row0 + m) * DMODEL;
  v8f acc[4] = {};
  for (int k = 0; k < 64; k += 4) {           // K 0..63: nf
    int kk = k + 2 * h;
    v2f a; a.x = arow[kk]; a.y = arow[kk + 1];
#pragma unroll
    for (int c = 0; c < 4; ++c) {
      v2f b;
      b.x = W[kk * DMODEL + c * 16 + m];
      b.y = W[(kk + 1) * DMODEL + c * 16 + m];
      acc[c] = wmma_f32(a, b, acc[c]);
    }
  }
  for (int k = 0; k < 64; k += 4) {           // K 64..127: agg
    int kk = k + 2 * h;
    v2f a; a.x = grow[kk]; a.y = grow[kk + 1];
#pragma unroll
    for (int c = 0; c < 4; ++c) {
      int wr = 64 + kk;
      v2f b;
      b.x = W[wr * DMODEL + c * 16 + m];
      b.y = W[(wr + 1) * DMODEL + c * 16 + m];
      acc[c] = wmma_f32(a, b, acc[c]);
    }
  }
#pragma unroll
  for (int c = 0; c < 4; ++c) {
    int col = c * 16 + m;
    float bv = bias[col];
#pragma unroll
    for (int r = 0; r < 8; ++r) {
      int row = row0 + r + 8 * h;
      float v = acc[c][r] + bv;
      v = v > 0.0f ? v : 0.0f;
      out[row * DMODEL + col] = nf[row * DMODEL + col] + v;
    }
  }
}

extern "C" void kernel_launch(void* const* d_in, const int* in_sizes, int n_in,
                              void* d_out, int out_size, void* d_ws,
                              size_t ws_size, hipStream_t stream) {
  const float* x   = (const float*)d_in[0];
  const float* ea  = (const float*)d_in[1];
  const float* Wne = (const float*)d_in[2];
  const float* bne = (const float*)d_in[3];
  const float* Wee = (const float*)d_in[4];
  const float* bee = (const float*)d_in[5];
  const float* Wem = (const float*)d_in[6];
  const float* bem = (const float*)d_in[7];
  const float* Wnm = (const float*)d_in[8];
  const float* bnm = (const float*)d_in[9];
  const int*   ei  = (const int*)d_in[10];
  const int* srcI = ei;
  const int* dstI = ei + N_EDGES;

  float* nf_out = (float*)d_out;                           // [N,64]
  float* ef_out = (float*)d_out + (size_t)N_NODES * DMODEL; // [E,64]
  float* nf_enc = (float*)d_ws;                            // [N,64] 12.8MB
  float* agg    = (float*)d_ws + (size_t)N_NODES * DMODEL; // [N,64] 12.8MB

  const int aggN = N_NODES * DMODEL;
  zero_kernel<<<(aggN + 255) / 256, 256, 0, stream>>>(agg, aggN);

  // 3125 node tiles, 8 waves/block
  node_enc_kernel<<<(N_NODES / 16 * 32 + 255) / 256, 256, 0, stream>>>(
      x, Wne, bne, nf_enc);

  // 50000 edge tiles, exactly 6250 blocks of 8 waves (uniform, no tail)
  edge_fused_kernel<<<N_EDGES / 16 / 8, 256, 0, stream>>>(
      ea, Wee, bee, nf_enc, Wem, bem, srcI, dstI, ef_out, agg);

  node_mlp_kernel<<<(N_NODES / 16 * 32 + 255) / 256, 256, 0, stream>>>(
      nf_enc, agg, Wnm, bnm, nf_out);
}